// Decoder_27960237097617
// MI455X (gfx1250) — compile-verified
//
#include <hip/hip_runtime.h>
#include <hip/hip_bf16.h>

// ---------------- problem constants ----------------
#define BB 64
#define PP 196
#define LL 20
#define ENC 2048
#define DEC 512
#define ATT 512
#define EMB 512
#define VOCAB 10000
#define NSTEP (LL - 1)

typedef __attribute__((ext_vector_type(2))) float v2f;
typedef __attribute__((ext_vector_type(8))) float v8f;

__device__ __forceinline__ float sigmoidf_(float x) { return 1.0f / (1.0f + __expf(-x)); }

// ---------------------------------------------------------------------------
// fp32 WMMA GEMM:  C[M,N] = epi( A[M,K] @ W[N,K]^T (+ A2@W2^T) + bias(+bias2) )
// Each wave computes a 64x32 output block: 4 M-tiles x 2 N-tiles of 16x16
// WMMA. Per 4-K chunk: 2 B-fragment loads + 4 A-fragment loads feed 8 WMMAs
// (10.7 FLOP / L2 byte). W is read once overall; A reads are halved vs the
// 64x16 version (A was the dominant L2 traffic for the skinny vocab/gates
// GEMMs). Odd N-tile counts (VOCAB/16 = 625) handled by clamping the dangling
// tile's loads and guarding its stores with a wave-uniform flag (EXEC stays
// all-ones through every WMMA, as required).
// EPI: 0 = none, 1 = tanh, 2 = sigmoid
// Requires: M%64==0, N%16==0, K%4==0.
// ---------------------------------------------------------------------------
template <int EPI>
__global__ __launch_bounds__(256) void gemm_wmma_f32(
    const float* __restrict__ A, int lda,
    const float* __restrict__ W, int ldw, int K,
    const float* __restrict__ A2, int lda2,
    const float* __restrict__ W2, int ldw2, int K2,
    const float* __restrict__ bias, const float* __restrict__ bias2,
    float* __restrict__ C, long ldc, int M, int N)
{
    const int wave     = (int)((blockIdx.x * blockDim.x + threadIdx.x) >> 5);
    const int tiles_n  = N >> 4;
    const int pairs_n  = (tiles_n + 1) >> 1;        // N-tile pairs per strip
    const int strips_m = M >> 6;                    // 64-row strips
    if (wave >= strips_m * pairs_n) return;         // wave-uniform exit: EXEC stays all-1

    const int sm   = wave / pairs_n;                // strip index (64 rows)
    const int pn   = wave % pairs_n;
    const int tn0  = pn * 2;
    const int tn1  = tn0 + 1;
    const bool has1 = (tn1 < tiles_n);              // wave-uniform
    const int tn1c = has1 ? tn1 : tn0;              // clamp for loads

    const int lane = (int)(threadIdx.x & 31);
    const int half = lane >> 4;                     // 0: K+0..1   1: K+2..3
    const int l15  = lane & 15;

    // A 16x4 fp32 layout: lanes 0-15 rows M0..15 (K,K+1), lanes 16-31 same rows (K+2,K+3)
    const float* ap = A + (size_t)(sm * 64 + l15) * lda + half * 2;
    const size_t arow16 = (size_t)16 * lda;
    // B 4x16 fp32 layout: lanes hold column n = l15 ; B[k][n] = W[n][k]
    const float* wp0 = W + (size_t)(tn0 * 16 + l15) * ldw + half * 2;
    const float* wp1 = W + (size_t)(tn1c * 16 + l15) * ldw + half * 2;

    v8f z = (v8f){0.f,0.f,0.f,0.f,0.f,0.f,0.f,0.f};
    v8f c00 = z, c10 = z, c20 = z, c30 = z;   // N-tile 0
    v8f c01 = z, c11 = z, c21 = z, c31 = z;   // N-tile 1

#pragma unroll 2
    for (int k = 0; k < K; k += 4) {
        v2f b0 = *(const v2f*)(wp0 + k);
        v2f b1 = *(const v2f*)(wp1 + k);
        v2f a0 = *(const v2f*)(ap + k);
        v2f a1 = *(const v2f*)(ap + arow16 + k);
        v2f a2 = *(const v2f*)(ap + 2 * arow16 + k);
        v2f a3 = *(const v2f*)(ap + 3 * arow16 + k);
        c00 = __builtin_amdgcn_wmma_f32_16x16x4_f32(false, a0, false, b0, (short)0, c00, false, false);
        c01 = __builtin_amdgcn_wmma_f32_16x16x4_f32(false, a0, false, b1, (short)0, c01, false, false);
        c10 = __builtin_amdgcn_wmma_f32_16x16x4_f32(false, a1, false, b0, (short)0, c10, false, false);
        c11 = __builtin_amdgcn_wmma_f32_16x16x4_f32(false, a1, false, b1, (short)0, c11, false, false);
        c20 = __builtin_amdgcn_wmma_f32_16x16x4_f32(false, a2, false, b0, (short)0, c20, false, false);
        c21 = __builtin_amdgcn_wmma_f32_16x16x4_f32(false, a2, false, b1, (short)0, c21, false, false);
        c30 = __builtin_amdgcn_wmma_f32_16x16x4_f32(false, a3, false, b0, (short)0, c30, false, false);
        c31 = __builtin_amdgcn_wmma_f32_16x16x4_f32(false, a3, false, b1, (short)0, c31, false, false);
    }

    if (A2 != nullptr) {  // second accumulation pass (gates = x@W_ih^T + h@W_hh^T)
        const float* ap2 = A2 + (size_t)(sm * 64 + l15) * lda2 + half * 2;
        const size_t arow16b = (size_t)16 * lda2;
        const float* wq0 = W2 + (size_t)(tn0 * 16 + l15) * ldw2 + half * 2;
        const float* wq1 = W2 + (size_t)(tn1c * 16 + l15) * ldw2 + half * 2;
#pragma unroll 2
        for (int k = 0; k < K2; k += 4) {
            v2f b0 = *(const v2f*)(wq0 + k);
            v2f b1 = *(const v2f*)(wq1 + k);
            v2f a0 = *(const v2f*)(ap2 + k);
            v2f a1 = *(const v2f*)(ap2 + arow16b + k);
            v2f a2 = *(const v2f*)(ap2 + 2 * arow16b + k);
            v2f a3 = *(const v2f*)(ap2 + 3 * arow16b + k);
            c00 = __builtin_amdgcn_wmma_f32_16x16x4_f32(false, a0, false, b0, (short)0, c00, false, false);
            c01 = __builtin_amdgcn_wmma_f32_16x16x4_f32(false, a0, false, b1, (short)0, c01, false, false);
            c10 = __builtin_amdgcn_wmma_f32_16x16x4_f32(false, a1, false, b0, (short)0, c10, false, false);
            c11 = __builtin_amdgcn_wmma_f32_16x16x4_f32(false, a1, false, b1, (short)0, c11, false, false);
            c20 = __builtin_amdgcn_wmma_f32_16x16x4_f32(false, a2, false, b0, (short)0, c20, false, false);
            c21 = __builtin_amdgcn_wmma_f32_16x16x4_f32(false, a2, false, b1, (short)0, c21, false, false);
            c30 = __builtin_amdgcn_wmma_f32_16x16x4_f32(false, a3, false, b0, (short)0, c30, false, false);
            c31 = __builtin_amdgcn_wmma_f32_16x16x4_f32(false, a3, false, b1, (short)0, c31, false, false);
        }
    }

    float bv0 = (bias  != nullptr) ? bias[tn0 * 16 + l15]  : 0.0f;
    float bv1 = (bias  != nullptr) ? bias[tn1c * 16 + l15] : 0.0f;
    if (bias2 != nullptr) { bv0 += bias2[tn0 * 16 + l15]; bv1 += bias2[tn1c * 16 + l15]; }

    // C/D layout per tile: VGPR r -> row (half*8 + r), col l15
    v8f acc0[4] = {c00, c10, c20, c30};
    v8f acc1[4] = {c01, c11, c21, c31};
#pragma unroll
    for (int mt = 0; mt < 4; ++mt) {
        float* crow = C + (size_t)(sm * 64 + mt * 16 + half * 8) * ldc + tn0 * 16 + l15;
#pragma unroll
        for (int r = 0; r < 8; ++r) {
            float v = acc0[mt][r] + bv0;
            if (EPI == 1) v = tanhf(v);
            else if (EPI == 2) v = sigmoidf_(v);
            crow[(size_t)r * ldc] = v;
        }
    }
    if (has1) {                                      // wave-uniform guard
#pragma unroll
        for (int mt = 0; mt < 4; ++mt) {
            float* crow = C + (size_t)(sm * 64 + mt * 16 + half * 8) * ldc + tn1 * 16 + l15;
#pragma unroll
            for (int r = 0; r < 8; ++r) {
                float v = acc1[mt][r] + bv1;
                if (EPI == 1) v = tanhf(v);
                else if (EPI == 2) v = sigmoidf_(v);
                crow[(size_t)r * ldc] = v;
            }
        }
    }
}

// ---------------------------------------------------------------------------
// avg[b][e] = mean_p img[b][p][e]
// ---------------------------------------------------------------------------
__global__ __launch_bounds__(256) void avg_kernel(const float* __restrict__ img,
                                                  float* __restrict__ avg)
{
    int i = (int)(blockIdx.x * blockDim.x + threadIdx.x);
    if (i >= BB * ENC) return;
    int b = i / ENC, e = i % ENC;
    const float* p = img + (size_t)b * PP * ENC + e;
    float s = 0.f;
#pragma unroll 4
    for (int pp = 0; pp < PP; ++pp) s += p[(size_t)pp * ENC];
    avg[i] = s * (1.0f / (float)PP);
}

// ---------------------------------------------------------------------------
// e[b,p] = sum_a tanh(enc_proj[b,p,a] + U_h[b,a]) * W_fin[a] + b_fin
// one wave per (b,p)
// ---------------------------------------------------------------------------
__global__ __launch_bounds__(256) void att_e_kernel(const float* __restrict__ enc_proj,
                                                    const float* __restrict__ U_h,
                                                    const float* __restrict__ W_fin,
                                                    const float* __restrict__ b_fin,
                                                    float* __restrict__ e_buf)
{
    int bp = (int)((blockIdx.x * blockDim.x + threadIdx.x) >> 5);
    if (bp >= BB * PP) return;
    int b    = bp / PP;
    int lane = (int)(threadIdx.x & 31);
    const float* ep = enc_proj + (size_t)bp * ATT;
    const float* uh = U_h + (size_t)b * ATT;
    float s = 0.f;
#pragma unroll 4
    for (int a = lane; a < ATT; a += 32) s += tanhf(ep[a] + uh[a]) * W_fin[a];
#pragma unroll
    for (int o = 16; o > 0; o >>= 1) s += __shfl_xor(s, o, 32);
    if (lane == 0) e_buf[bp] = s + b_fin[0];
}

// ---------------------------------------------------------------------------
// per-b: alpha = softmax(e[b,:]);  write alphas[b,t,:];
//        context[b,:] = sum_p alpha[p] * img[b,p,:]
// one block (256 threads) per b
// ---------------------------------------------------------------------------
__global__ __launch_bounds__(256) void softmax_ctx_kernel(const float* __restrict__ e_buf,
                                                          const float* __restrict__ img,
                                                          float* __restrict__ alpha_out,
                                                          long alpha_ld,
                                                          float* __restrict__ context)
{
    __shared__ float salpha[PP];
    __shared__ float swred[8];
    int b = (int)blockIdx.x;
    int tid = (int)threadIdx.x, lane = tid & 31, wid = tid >> 5;

    float v = (tid < PP) ? e_buf[b * PP + tid] : -3.4e38f;
    float m = v;
#pragma unroll
    for (int o = 16; o > 0; o >>= 1) m = fmaxf(m, __shfl_xor(m, o, 32));
    if (lane == 0) swred[wid] = m;
    __syncthreads();
    float bm = swred[0];
#pragma unroll
    for (int w = 1; w < 8; ++w) bm = fmaxf(bm, swred[w]);

    float ex = (tid < PP) ? __expf(v - bm) : 0.f;
    float s = ex;
#pragma unroll
    for (int o = 16; o > 0; o >>= 1) s += __shfl_xor(s, o, 32);
    __syncthreads();
    if (lane == 0) swred[wid] = s;
    __syncthreads();
    float bs = 0.f;
#pragma unroll
    for (int w = 0; w < 8; ++w) bs += swred[w];

    float alpha = ex / bs;
    if (tid < PP) {
        salpha[tid] = alpha;
        alpha_out[(size_t)b * alpha_ld + tid] = alpha;
    }
    __syncthreads();

    const float* ip = img + (size_t)b * PP * ENC;
    for (int c0 = tid; c0 < ENC; c0 += 256) {
        float acc = 0.f;
#pragma unroll 4
        for (int p = 0; p < PP; ++p) acc += salpha[p] * ip[(size_t)p * ENC + c0];
        context[(size_t)b * ENC + c0] = acc;
    }
}

// ---------------------------------------------------------------------------
// x[b, :] = [ emb[captions[b][t]] , gate_s[b,:] * context[b,:] ]
// ---------------------------------------------------------------------------
__global__ __launch_bounds__(256) void build_x_kernel(const float* __restrict__ emb,
                                                      const int* __restrict__ captions,
                                                      int t,
                                                      const float* __restrict__ gate_s,
                                                      const float* __restrict__ context,
                                                      float* __restrict__ x)
{
    int i = (int)(blockIdx.x * blockDim.x + threadIdx.x);
    if (i >= BB * (EMB + ENC)) return;
    int b = i / (EMB + ENC), j = i % (EMB + ENC);
    float v;
    if (j < EMB) {
        int tok = captions[b * LL + t];
        v = emb[(size_t)tok * EMB + j];
    } else {
        int jj = j - EMB;
        v = gate_s[(size_t)b * ENC + jj] * context[(size_t)b * ENC + jj];
    }
    x[(size_t)b * (EMB + ENC) + j] = v;
}

// ---------------------------------------------------------------------------
// LSTM pointwise: gates[b, 0:4*DEC] -> (h,c) updated in place
// ---------------------------------------------------------------------------
__global__ __launch_bounds__(256) void lstm_update_kernel(const float* __restrict__ gates,
                                                          float* __restrict__ h,
                                                          float* __restrict__ c)
{
    int i = (int)(blockIdx.x * blockDim.x + threadIdx.x);
    if (i >= BB * DEC) return;
    int b = i / DEC, j = i % DEC;
    const float* g = gates + (size_t)b * 4 * DEC;
    float ig = sigmoidf_(g[j]);
    float fg = sigmoidf_(g[DEC + j]);
    float gg = tanhf(g[2 * DEC + j]);
    float og = sigmoidf_(g[3 * DEC + j]);
    float cn = fg * c[i] + ig * gg;
    c[i] = cn;
    h[i] = og * tanhf(cn);
}

// ---------------------------------------------------------------------------
// host-side launch helpers
// ---------------------------------------------------------------------------
static inline int gemm_blocks(int M, int N) {
    int tiles_n = N >> 4;
    int waves = (M >> 6) * ((tiles_n + 1) >> 1);   // one wave per 64x32 block
    return (waves + 7) / 8;                        // 8 waves / 256-thread block
}

template <int EPI>
static void launch_gemm(const float* A, int lda, const float* W, int ldw, int K,
                        const float* bias, float* C, long ldc, int M, int N,
                        hipStream_t stream,
                        const float* A2 = nullptr, int lda2 = 0,
                        const float* W2 = nullptr, int ldw2 = 0, int K2 = 0,
                        const float* bias2 = nullptr)
{
    gemm_wmma_f32<EPI><<<gemm_blocks(M, N), 256, 0, stream>>>(
        A, lda, W, ldw, K, A2, lda2, W2, ldw2, K2, bias, bias2, C, ldc, M, N);
}

extern "C" void kernel_launch(void* const* d_in, const int* in_sizes, int n_in,
                              void* d_out, int out_size, void* d_ws, size_t ws_size,
                              hipStream_t stream)
{
    // inputs in setup_inputs() dict order
    const float* img       = (const float*)d_in[0];
    const int*   captions  = (const int*)  d_in[1];
    const float* W_init_h  = (const float*)d_in[2];
    const float* b_init_h  = (const float*)d_in[3];
    const float* W_init_c  = (const float*)d_in[4];
    const float* b_init_c  = (const float*)d_in[5];
    const float* W_dec     = (const float*)d_in[6];
    const float* b_dec     = (const float*)d_in[7];
    const float* W_enc     = (const float*)d_in[8];
    const float* b_enc     = (const float*)d_in[9];
    const float* W_fin     = (const float*)d_in[10];
    const float* b_fin     = (const float*)d_in[11];
    const float* W_gate    = (const float*)d_in[12];
    const float* b_gate    = (const float*)d_in[13];
    const float* W_out     = (const float*)d_in[14];
    const float* b_out     = (const float*)d_in[15];
    const float* emb       = (const float*)d_in[16];
    const float* W_ih      = (const float*)d_in[17];
    const float* b_ih      = (const float*)d_in[18];
    const float* W_hh      = (const float*)d_in[19];
    const float* b_hh      = (const float*)d_in[20];

    // workspace layout (floats)
    float* ws = (float*)d_ws;
    float* enc_proj = ws;                                   // 12544*512
    float* avg      = enc_proj + (size_t)BB * PP * ATT;     // 64*2048
    float* h        = avg      + (size_t)BB * ENC;          // 64*512
    float* c        = h        + (size_t)BB * DEC;          // 64*512
    float* uh       = c        + (size_t)BB * DEC;          // 64*512
    float* e_buf    = uh       + (size_t)BB * ATT;          // 64*196
    float* context  = e_buf    + (size_t)BB * PP;           // 64*2048
    float* gate_s   = context  + (size_t)BB * ENC;          // 64*2048
    float* x        = gate_s   + (size_t)BB * ENC;          // 64*2560
    float* gates    = x        + (size_t)BB * (EMB + ENC);  // 64*2048

    float* preds  = (float*)d_out;                              // [B, 19, VOCAB]
    float* alphas = preds + (size_t)BB * NSTEP * VOCAB;         // [B, 19, P]

    // ---- prologue ----
    avg_kernel<<<(BB * ENC + 255) / 256, 256, 0, stream>>>(img, avg);

    // h0 = tanh(avg @ W_init_h^T + b) ; c0 likewise
    launch_gemm<1>(avg, ENC, W_init_h, ENC, ENC, b_init_h, h, DEC, BB, DEC, stream);
    launch_gemm<1>(avg, ENC, W_init_c, ENC, ENC, b_init_c, c, DEC, BB, DEC, stream);

    // enc_proj = img(12544x2048) @ W_enc^T + b_enc
    launch_gemm<0>(img, ENC, W_enc, ENC, ENC, b_enc, enc_proj, ATT, BB * PP, ATT, stream);

    // ---- 19-step recurrence ----
    for (int t = 0; t < NSTEP; ++t) {
        // U_h = h @ W_dec^T + b_dec
        launch_gemm<0>(h, DEC, W_dec, DEC, DEC, b_dec, uh, ATT, BB, ATT, stream);

        // attention energies
        att_e_kernel<<<(BB * PP + 7) / 8, 256, 0, stream>>>(enc_proj, uh, W_fin, b_fin, e_buf);

        // softmax + context; alphas written straight into output tensor
        softmax_ctx_kernel<<<BB, 256, 0, stream>>>(
            e_buf, img, alphas + (size_t)t * PP, (long)NSTEP * PP, context);

        // gate = sigmoid(h @ W_gate^T + b_gate)
        launch_gemm<2>(h, DEC, W_gate, DEC, DEC, b_gate, gate_s, ENC, BB, ENC, stream);

        // x = [emb[cap[:,t]], gate * context]
        build_x_kernel<<<(BB * (EMB + ENC) + 255) / 256, 256, 0, stream>>>(
            emb, captions, t, gate_s, context, x);

        // gates = x @ W_ih^T + h @ W_hh^T + b_ih + b_hh   (dual-K accumulation)
        launch_gemm<0>(x, EMB + ENC, W_ih, EMB + ENC, EMB + ENC, b_ih,
                       gates, 4 * DEC, BB, 4 * DEC, stream,
                       h, DEC, W_hh, DEC, DEC, b_hh);

        // LSTM pointwise update (h, c in place)
        lstm_update_kernel<<<(BB * DEC + 255) / 256, 256, 0, stream>>>(gates, h, c);

        // preds[:, t, :] = h @ W_out^T + b_out
        launch_gemm<0>(h, DEC, W_out, DEC, DEC, b_out,
                       preds + (size_t)t * VOCAB, (long)NSTEP * VOCAB, BB, VOCAB, stream);
    }
}